// GHM_Loss_6073083756771
// MI455X (gfx1250) — compile-verified
//
#include <hip/hip_runtime.h>
#include <hip/hip_bf16.h>

#define GHM_BINS 10

typedef __attribute__((ext_vector_type(2))) float v2f;
typedef __attribute__((ext_vector_type(8))) float v8f;

// ---------------------------------------------------------------------------
// Per-element GHM work: one exp, one rcp, one log (all hardware trans ops),
// two conflict-free ds_add_f32 LDS atomics into this thread's private bin slot.
// ---------------------------------------------------------------------------
__device__ __forceinline__ void ghm_elem(float x, float t,
                                         float* __restrict__ lcnt,
                                         float* __restrict__ lsum,
                                         int tid)
{
    float e    = __expf(-__builtin_fabsf(x));          // e = exp(-|x|)
    float onep = 1.0f + e;
    float r    = __builtin_amdgcn_rcpf(onep);          // 1/(1+e)
    float sig  = (x >= 0.0f) ? r : e * r;              // stable sigmoid
    int   bin  = (int)(__builtin_fabsf(sig - t) * 9.9999f);   // floor(g*(10-1e-4))
    // bce = max(x,0) - x*t + log1p(e), and log1p(e) == log(onep): reuse onep
    float bce  = __builtin_fmaf(-x, t, __builtin_fmaxf(x, 0.0f)) + __logf(onep);

    __hip_atomic_fetch_add(&lcnt[bin * 256 + tid], 1.0f,
                           __ATOMIC_RELAXED, __HIP_MEMORY_SCOPE_WORKGROUP);
    __hip_atomic_fetch_add(&lsum[bin * 256 + tid], bce,
                           __ATOMIC_RELAXED, __HIP_MEMORY_SCOPE_WORKGROUP);
}

// ---------------------------------------------------------------------------
// Zero the 20-float global accumulator (ws is poisoned by the harness and not
// re-poisoned between replays, so this must run every launch).
// ---------------------------------------------------------------------------
__global__ void ghm_zero(float* __restrict__ ws)
{
    if (threadIdx.x < 2 * GHM_BINS) ws[threadIdx.x] = 0.0f;
}

// ---------------------------------------------------------------------------
// Main fused pass: stream x/target once (b128 loads), build per-bin
// (count, bce-sum) in per-thread-slot LDS histograms, flush via global
// float atomics.  ws layout: [0..9] = counts, [10..19] = bce sums.
// ---------------------------------------------------------------------------
__global__ __launch_bounds__(256) void ghm_main(const float4* __restrict__ x4,
                                                const float4* __restrict__ t4,
                                                long long n4, long long n,
                                                float* __restrict__ gacc)
{
    __shared__ __align__(16) float lcnt[GHM_BINS * 256];
    __shared__ __align__(16) float lsum[GHM_BINS * 256];
    const int tid = threadIdx.x;

#pragma unroll
    for (int b = 0; b < GHM_BINS; ++b) {
        lcnt[b * 256 + tid] = 0.0f;
        lsum[b * 256 + tid] = 0.0f;
    }
    __syncthreads();

    const long long stride = (long long)gridDim.x * 256;
    for (long long i = (long long)blockIdx.x * 256 + tid; i < n4; i += stride) {
        float4 xv = x4[i];
        float4 tv = t4[i];
        ghm_elem(xv.x, tv.x, lcnt, lsum, tid);
        ghm_elem(xv.y, tv.y, lcnt, lsum, tid);
        ghm_elem(xv.z, tv.z, lcnt, lsum, tid);
        ghm_elem(xv.w, tv.w, lcnt, lsum, tid);
    }

    // Scalar tail (n not a multiple of 4) handled by block 0.
    if (blockIdx.x == 0) {
        long long idx = (n4 << 2) + tid;
        if (idx < n) {
            const float* xs = (const float*)x4;
            const float* ts = (const float*)t4;
            ghm_elem(xs[idx], ts[idx], lcnt, lsum, tid);
        }
    }
    __syncthreads();

    // Block reduction: 20 threads each collapse one 256-slot bin (b128 LDS
    // loads), then one global float atomic per bin per block.
    if (tid < 2 * GHM_BINS) {
        const bool    isSum = tid >= GHM_BINS;
        const int     b     = isSum ? tid - GHM_BINS : tid;
        const float4* src   = (const float4*)(isSum ? &lsum[b * 256] : &lcnt[b * 256]);
        float4 s = make_float4(0.0f, 0.0f, 0.0f, 0.0f);
#pragma unroll 4
        for (int j = 0; j < 64; ++j) {
            float4 v = src[j];
            s.x += v.x; s.y += v.y; s.z += v.z; s.w += v.w;
        }
        float tot = (s.x + s.y) + (s.z + s.w);
        __hip_atomic_fetch_add(&gacc[tid], tot,
                               __ATOMIC_RELAXED, __HIP_MEMORY_SCOPE_AGENT);
    }
}

// ---------------------------------------------------------------------------
// Finalize: result = sum_b S[b] / max(cnt[b]*nonempty, 1e-4)   (N cancels).
// The 16-term dot is evaluated on the matrix unit with
// V_WMMA_F32_16X16X4_F32:  A[m,k] = v[4m+k] (documented 16x4 f32 A layout),
// B = all-ones (B layout immaterial), so D[m,n] = sum_k v[4m+k]; lane 0's
// four accumulator VGPRs (M=0..3, N=0) sum to the answer.
// Launched with exactly 32 threads; no divergence before the WMMA (EXEC all-1).
// ---------------------------------------------------------------------------
__global__ __launch_bounds__(32) void ghm_final(const float* __restrict__ ws,
                                                float* __restrict__ out)
{
    const int lane = threadIdx.x;

    float cnt[GHM_BINS], sum[GHM_BINS];
    float nonempty = 0.0f;
#pragma unroll
    for (int b = 0; b < GHM_BINS; ++b) {
        cnt[b] = ws[b];
        sum[b] = ws[GHM_BINS + b];
        nonempty += (cnt[b] > 0.0f) ? 1.0f : 0.0f;
    }

    float v[16];
#pragma unroll
    for (int b = 0; b < 16; ++b) {
        if (b < GHM_BINS) {
            float gd = __builtin_fmaxf(cnt[b] * nonempty, 1.0e-4f);
            v[b] = sum[b] / gd;
        } else {
            v[b] = 0.0f;
        }
    }

    // A-matrix 16x4 f32 layout: lanes 0-15 hold rows M=0..15 with K=0,1 in the
    // two A VGPRs; lanes 16-31 hold the same rows with K=2,3.
    const int m  = lane & 15;
    const int kb = (lane < 16) ? 0 : 2;
    const int j0 = 4 * m + kb;            // j0 >= 16 for m >= 4 -> stays zero
    float ax = 0.0f, ay = 0.0f;
#pragma unroll
    for (int q = 0; q < 16; ++q) {        // branch-free select (cndmask chain)
        ax = (j0 == q)     ? v[q] : ax;
        ay = (j0 + 1 == q) ? v[q] : ay;
    }

    v2f a;  a.x = ax;   a.y = ay;
    v2f bm; bm.x = 1.0f; bm.y = 1.0f;     // B = ones
    v8f c = {};
    c = __builtin_amdgcn_wmma_f32_16x16x4_f32(
            /*neg_a=*/false, a, /*neg_b=*/false, bm,
            /*c_mod=*/(short)0, c, /*reuse_a=*/false, /*reuse_b=*/false);

    float total = (c[0] + c[1]) + (c[2] + c[3]);   // D[0..3, 0] on lane 0
    if (lane == 0) out[0] = total;
}

// ---------------------------------------------------------------------------
extern "C" void kernel_launch(void* const* d_in, const int* in_sizes, int n_in,
                              void* d_out, int out_size, void* d_ws, size_t ws_size,
                              hipStream_t stream)
{
    const float* x = (const float*)d_in[0];
    const float* t = (const float*)d_in[1];
    float* out = (float*)d_out;
    float* ws  = (float*)d_ws;   // 20 floats: [0..9] counts, [10..19] bce sums

    const long long n  = (long long)in_sizes[0];
    const long long n4 = n >> 2;

    int blocks = 1024;                       // 8192 wave32s; ~16 float4/thread
    long long maxBlocks = (n4 + 255) / 256;
    if (maxBlocks < 1) maxBlocks = 1;
    if ((long long)blocks > maxBlocks) blocks = (int)maxBlocks;

    ghm_zero <<<1, 32, 0, stream>>>(ws);
    ghm_main <<<blocks, 256, 0, stream>>>((const float4*)x, (const float4*)t,
                                          n4, n, ws);
    ghm_final<<<1, 32, 0, stream>>>(ws, out);
}